// MLP_69363721831001
// MI455X (gfx1250) — compile-verified
//
#include <hip/hip_runtime.h>
#include <math.h>

// ---------------------------------------------------------------------------
// Per-cell tiny MLP + gated pooling, fp32 WMMA path for gfx1250 (MI455X).
//   X[64,20000,64] -> h1=relu(X*W1+b1)[...,16] -> h2=relu(h1*W2+b2)[...,8]
//   w=sigmoid(h2*Ww+bw); pooled[b,8]=sum_n w*h2 / sum_n w; out=pooled*W4+b4
// Memory-bound on streaming X (328MB @ 23.3TB/s ~ 14us); fp32 WMMA keeps us
// on that roofline with no precision loss.
// ---------------------------------------------------------------------------

typedef __attribute__((ext_vector_type(2))) float v2f;
typedef __attribute__((ext_vector_type(8))) float v8f;

#define N_INPUTS 64
#define H1DIM    16
#define H2DIM    8
#define BATCH    64
#define NCELLS   20000
#define CHUNK    16
#define NCHUNKS  (NCELLS / CHUNK)      // 1250 (exact)
#define WAVES_PB 8                      // 256 threads = 8 wave32
#define BLOCKS_PER_BATCH 32

// ---- zero the [64][9] accumulator workspace (graph-capture safe) ----------
__global__ void mlp_zero_ws(float* ws) {
    int i = blockIdx.x * blockDim.x + threadIdx.x;
    if (i < BATCH * 9) ws[i] = 0.0f;
}

// ---- main streaming kernel ------------------------------------------------
__global__ __launch_bounds__(256) void mlp_pool_kernel(
    const float* __restrict__ X,
    const float* __restrict__ W1, const float* __restrict__ b1,
    const float* __restrict__ W2, const float* __restrict__ b2,
    const float* __restrict__ Ww, const float* __restrict__ bw,
    float* __restrict__ ws)
{
    __shared__ float lds_h1[WAVES_PB][16 * 16];   // per-wave h1 tile
    __shared__ float lds_h2[WAVES_PB][16 * 8];    // per-wave h2 tile
    __shared__ float blockAcc[9];                 // [0..7]=sum(w*h2), [8]=sum(w)

    const int tid  = threadIdx.x;
    const int wave = tid >> 5;
    const int lane = tid & 31;
    const int n    = lane & 15;   // column (B/C layout) or row M (A layout)
    const int kh   = lane >> 4;   // half selector (ISA half-row interleave)
    const int b    = blockIdx.y;

    if (tid < 9) blockAcc[tid] = 0.0f;
    __syncthreads();

    // ---- preload W1 B-fragments (B layout: VGPR0 holds K rows {4k, 4k+2},
    //      VGPR1 holds {4k+1, 4k+3}; row striped across lanes) --------------
    v2f w1f[16];
#pragma unroll
    for (int k = 0; k < 16; ++k) {
        w1f[k].x = W1[(4 * k + 2 * kh + 0) * H1DIM + n];
        w1f[k].y = W1[(4 * k + 2 * kh + 1) * H1DIM + n];
    }
    // ---- W2 zero-padded to 16x16, as 4 B-fragments -------------------------
    v2f w2f[4];
#pragma unroll
    for (int j = 0; j < 4; ++j) {
        int r0 = 4 * j + 2 * kh;
        w2f[j].x = (n < H2DIM) ? W2[(r0 + 0) * H2DIM + n] : 0.0f;
        w2f[j].y = (n < H2DIM) ? W2[(r0 + 1) * H2DIM + n] : 0.0f;
    }
    const float b1n = b1[n];
    const float b2n = (n < H2DIM) ? b2[n] : 0.0f;
    float wwreg[8];
#pragma unroll
    for (int k = 0; k < 8; ++k) wwreg[k] = Ww[k];
    const float bwreg = bw[0];

    float accw = 0.0f;
    float acc[8];
#pragma unroll
    for (int k = 0; k < 8; ++k) acc[k] = 0.0f;

    for (int chunk = blockIdx.x * WAVES_PB + wave; chunk < NCHUNKS;
         chunk += gridDim.x * WAVES_PB)
    {
        // Lane's cell row (A layout: lanes 0..15 hold M=0..15, both halves)
        const float* xrow =
            X + ((size_t)b * NCELLS + (size_t)chunk * CHUNK + n) * N_INPUTS;

        // ---- A fragments for layer 1: float2 {K=4k+2kh, 4k+1+2kh} ---------
        v2f a[16];
#pragma unroll
        for (int k = 0; k < 16; ++k)
            a[k] = *(const v2f*)(xrow + 4 * k + 2 * kh);

        // ---- layer 1: C init = b1 (bias folded), 16 chained f32 WMMAs -----
        v8f c;
#pragma unroll
        for (int r = 0; r < 8; ++r) c[r] = b1n;
#pragma unroll
        for (int k = 0; k < 16; ++k)
            c = __builtin_amdgcn_wmma_f32_16x16x4_f32(
                    false, a[k], false, w1f[k], (short)0, c, false, false);

        // ---- relu(h1) -> LDS (C layout: lane holds col n, rows r+8*kh) ----
#pragma unroll
        for (int r = 0; r < 8; ++r)
            lds_h1[wave][(r + 8 * kh) * 16 + n] = fmaxf(c[r], 0.0f);

        __builtin_amdgcn_wave_barrier();   // order intra-wave DS store->load

        // ---- layer 2: re-read h1 in A layout, 4 WMMAs vs padded W2 --------
        v8f c2;
#pragma unroll
        for (int r = 0; r < 8; ++r) c2[r] = b2n;
#pragma unroll
        for (int j = 0; j < 4; ++j) {
            v2f a2 = *(const v2f*)&lds_h1[wave][n * 16 + 4 * j + 2 * kh];
            c2 = __builtin_amdgcn_wmma_f32_16x16x4_f32(
                     false, a2, false, w2f[j], (short)0, c2, false, false);
        }

        // ---- relu(h2) cols 0..7 -> LDS ------------------------------------
        if (n < H2DIM) {
#pragma unroll
            for (int r = 0; r < 8; ++r)
                lds_h2[wave][(r + 8 * kh) * 8 + n] = fmaxf(c2[r], 0.0f);
        }

        __builtin_amdgcn_wave_barrier();

        // ---- gate + pooled accumulation: lanes 0..15 each own one cell ----
        if (lane < 16) {
            float h2v[8];
            float t = bwreg;
#pragma unroll
            for (int k = 0; k < 8; ++k) {
                h2v[k] = lds_h2[wave][lane * 8 + k];
                t += h2v[k] * wwreg[k];
            }
            float w = 1.0f / (1.0f + __expf(-t));
            accw += w;
#pragma unroll
            for (int k = 0; k < 8; ++k) acc[k] += w * h2v[k];
        }
        // reconvergence restores EXEC=all-ones before next iteration's WMMAs
    }

    // ---- block reduction: LDS float atomics, then one global atomic -------
    if (lane < 16) {
        atomicAdd(&blockAcc[8], accw);
#pragma unroll
        for (int k = 0; k < 8; ++k) atomicAdd(&blockAcc[k], acc[k]);
    }
    __syncthreads();
    if (tid < 9) atomicAdd(&ws[b * 9 + tid], blockAcc[tid]);
}

// ---- head: pooled = sum(w*h2)/sum(w); out = pooled @ W4 + b4 --------------
__global__ void mlp_head_kernel(const float* __restrict__ ws,
                                const float* __restrict__ W4,
                                const float* __restrict__ b4,
                                float* __restrict__ out)
{
    int bi = threadIdx.x;
    if (bi < BATCH) {
        const float* a = ws + bi * 9;
        float inv = 1.0f / a[8];
        float o0 = b4[0], o1 = b4[1];
#pragma unroll
        for (int k = 0; k < 8; ++k) {
            float p = a[k] * inv;
            o0 += p * W4[k * 2 + 0];
            o1 += p * W4[k * 2 + 1];
        }
        out[bi * 2 + 0] = o0;
        out[bi * 2 + 1] = o1;
    }
}

extern "C" void kernel_launch(void* const* d_in, const int* in_sizes, int n_in,
                              void* d_out, int out_size, void* d_ws, size_t ws_size,
                              hipStream_t stream)
{
    const float* X  = (const float*)d_in[0];
    const float* W1 = (const float*)d_in[1];
    const float* b1 = (const float*)d_in[2];
    const float* W2 = (const float*)d_in[3];
    const float* b2 = (const float*)d_in[4];
    const float* Ww = (const float*)d_in[5];
    const float* bw = (const float*)d_in[6];
    const float* W4 = (const float*)d_in[7];
    const float* b4 = (const float*)d_in[8];
    float* out = (float*)d_out;
    float* ws  = (float*)d_ws;   // needs 64*9*4 = 2304 bytes

    mlp_zero_ws<<<dim3((BATCH * 9 + 255) / 256), dim3(256), 0, stream>>>(ws);
    mlp_pool_kernel<<<dim3(BLOCKS_PER_BATCH, BATCH), dim3(256), 0, stream>>>(
        X, W1, b1, W2, b2, Ww, bw, ws);
    mlp_head_kernel<<<dim3(1), dim3(64), 0, stream>>>(ws, W4, b4, out);
}